// GE2ELoss_23948737642975
// MI455X (gfx1250) — compile-verified
//
#include <hip/hip_runtime.h>
#include <hip/hip_bf16.h>
#include <math.h>

typedef __attribute__((ext_vector_type(2))) float v2f;
typedef __attribute__((ext_vector_type(8))) float v8f;

#define N_SPK 1024
#define M_UTT 32
#define D_EMB 256
#define N_ROWS (N_SPK * M_UTT)   // 32768
#define EPS 1e-8f

// ---------------------------------------------------------------------------
// Kernel 1a: per-speaker centroid + normalized centroid (cn).
// 1024 blocks x 256 threads (thread d handles dim d).
// ---------------------------------------------------------------------------
__global__ __launch_bounds__(256) void k_centroid(const float* __restrict__ emb,
                                                  float* __restrict__ cent,
                                                  float* __restrict__ cn) {
    const int n = blockIdx.x;
    const int d = threadIdx.x;                     // 0..255
    const float* base = emb + (size_t)n * M_UTT * D_EMB + d;
    float s = 0.f;
#pragma unroll
    for (int m = 0; m < M_UTT; ++m) s += base[m * D_EMB];
    const float c = s * (1.0f / (float)M_UTT);
    cent[n * D_EMB + d] = c;

    __shared__ float red[256];
    red[d] = c * c;
    __syncthreads();
    for (int off = 128; off > 0; off >>= 1) {
        if (d < off) red[d] += red[d + off];
        __syncthreads();
    }
    const float inv = 1.0f / fmaxf(sqrtf(red[0]), EPS);
    cn[n * D_EMB + d] = c * inv;
}

// ---------------------------------------------------------------------------
// Kernel 1b: per-utterance row stats: inverse embedding norm and
// leave-one-out cosine similarity. 32768 blocks x 256 threads.
// ---------------------------------------------------------------------------
__global__ __launch_bounds__(256) void k_rowstats(const float* __restrict__ emb,
                                                  const float* __restrict__ cent,
                                                  float* __restrict__ loo_sim,
                                                  float* __restrict__ inv_e) {
    const int row = blockIdx.x;                    // 0..32767
    const int d   = threadIdx.x;
    const int spk = row >> 5;                      // row / 32
    const float e = emb[(size_t)row * D_EMB + d];
    const float c = cent[spk * D_EMB + d];
    const float loo = (c * (float)M_UTT - e) * (1.0f / (float)(M_UTT - 1));

    __shared__ float r0[256], r1[256], r2[256];
    r0[d] = e * e;
    r1[d] = loo * loo;
    r2[d] = e * loo;
    __syncthreads();
    for (int off = 128; off > 0; off >>= 1) {
        if (d < off) { r0[d] += r0[d + off]; r1[d] += r1[d + off]; r2[d] += r2[d + off]; }
        __syncthreads();
    }
    if (d == 0) {
        const float ie = 1.0f / fmaxf(sqrtf(r0[0]), EPS);
        const float il = 1.0f / fmaxf(sqrtf(r1[0]), EPS);
        inv_e[row]   = ie;
        loo_sim[row] = r2[0] * ie * il;
    }
}

// ---------------------------------------------------------------------------
// Kernel 2: fused GEMM (WMMA f32 16x16x4) + diagonal fix + logsumexp.
// One block = 16 rows (all within one speaker). 8 waves sweep the 64
// column tiles of the 1024 centroids. A tile is normalized into LDS once.
// Diagonal replacement is branchless: the 8 leave-one-out sims per lane are
// ct-loop invariant, preloaded into registers, and merged via v_cndmask.
// ---------------------------------------------------------------------------
__global__ __launch_bounds__(256) void k_ge2e_main(const float* __restrict__ emb,
                                                   const float* __restrict__ cn,
                                                   const float* __restrict__ loo_sim,
                                                   const float* __restrict__ inv_e,
                                                   const float* __restrict__ wp,
                                                   const float* __restrict__ bp,
                                                   float* __restrict__ partial) {
    __shared__ __align__(16) float As[16 * D_EMB];   // 16 KB normalized A rows
    __shared__ float wsum[8][16];                    // per-wave, per-row partial sums

    const int tid  = threadIdx.x;
    const int wave = tid >> 5;
    const int lane = tid & 31;
    const int half = lane >> 4;          // K-half selector for A/B fragments
    const int l16  = lane & 15;
    const int rowBase = blockIdx.x * 16; // 16 rows, all in speaker rowBase/32
    const int spk = rowBase >> 5;

    // Stage A: 16 rows x 256 dims = 1024 float4; 256 threads -> 4 each.
    for (int i = tid; i < 16 * 64; i += 256) {
        const int r  = i >> 6;
        const int c4 = i & 63;
        float4 v = ((const float4*)(emb + (size_t)(rowBase + r) * D_EMB))[c4];
        const float s = inv_e[rowBase + r];
        v.x *= s; v.y *= s; v.z *= s; v.w *= s;
        ((float4*)(As + r * D_EMB))[c4] = v;
    }
    __syncthreads();

    const float w_c   = fmaxf(wp[0], 0.01f);
    const float bias  = bp[0];
    const float shift = w_c + bias;      // max possible scaled value (cos <= 1)

    // Preload this lane's 8 leave-one-out sims (row = v + 8*half, fixed per lane).
    float loo_v[8];
#pragma unroll
    for (int v = 0; v < 8; ++v) loo_v[v] = loo_sim[rowBase + v + 8 * half];

    float rs[8];
#pragma unroll
    for (int v = 0; v < 8; ++v) rs[v] = 0.f;

    const float* Arow = As + l16 * D_EMB + 2 * half;

    for (int ct = wave; ct < 64; ct += 8) {          // 8 column tiles per wave
        const int colBase = ct * 16;
        const float* Brow = cn + (size_t)(colBase + l16) * D_EMB + 2 * half;

        v8f acc0 = {};
        v8f acc1 = {};
#pragma unroll 4
        for (int k0 = 0; k0 < D_EMB; k0 += 8) {
            v2f a0, b0, a1, b1;
            a0.x = Arow[k0];     a0.y = Arow[k0 + 1];
            b0.x = Brow[k0];     b0.y = Brow[k0 + 1];
            a1.x = Arow[k0 + 4]; a1.y = Arow[k0 + 5];
            b1.x = Brow[k0 + 4]; b1.y = Brow[k0 + 5];
            acc0 = __builtin_amdgcn_wmma_f32_16x16x4_f32(false, a0, false, b0,
                                                         (short)0, acc0, false, false);
            acc1 = __builtin_amdgcn_wmma_f32_16x16x4_f32(false, a1, false, b1,
                                                         (short)0, acc1, false, false);
        }

        const bool diag = (colBase + l16) == spk;    // this lane hits the diagonal?
#pragma unroll
        for (int v = 0; v < 8; ++v) {
            float s = acc0[v] + acc1[v];             // S[row, colg]
            s = diag ? loo_v[v] : s;                 // branchless v_cndmask
            rs[v] += __expf(w_c * s + bias - shift);
        }
    }

    // Reduce each row's partial over the 16 lanes of its half-wave.
#pragma unroll
    for (int v = 0; v < 8; ++v) {
        float x = rs[v];
        for (int m = 1; m < 16; m <<= 1) x += __shfl_xor(x, m, 32);
        rs[v] = x;
    }
    if (l16 == 0) {
#pragma unroll
        for (int v = 0; v < 8; ++v) wsum[wave][half * 8 + v] = rs[v];
    }
    __syncthreads();

    // Combine the 8 waves, finish logsumexp, accumulate block loss.
    if (tid < 16) {
        float s = 0.f;
#pragma unroll
        for (int wv = 0; wv < 8; ++wv) s += wsum[wv][tid];
        const float lse = shift + __logf(s);
        const float pos = w_c * loo_sim[rowBase + tid] + bias;
        float contrib = lse - pos;
        for (int m = 1; m < 16; m <<= 1) contrib += __shfl_xor(contrib, m, 32);
        if (tid == 0) partial[blockIdx.x] = contrib;
    }
}

// ---------------------------------------------------------------------------
// Kernel 3: deterministic final reduction of 2048 block partials.
// ---------------------------------------------------------------------------
__global__ __launch_bounds__(256) void k_final(const float* __restrict__ partial,
                                               float* __restrict__ out) {
    __shared__ float red[256];
    const int t = threadIdx.x;
    float s = 0.f;
    for (int i = t; i < 2048; i += 256) s += partial[i];
    red[t] = s;
    __syncthreads();
    for (int off = 128; off > 0; off >>= 1) {
        if (t < off) red[t] += red[t + off];
        __syncthreads();
    }
    if (t == 0) out[0] = red[0] * (1.0f / (float)N_ROWS);
}

extern "C" void kernel_launch(void* const* d_in, const int* in_sizes, int n_in,
                              void* d_out, int out_size, void* d_ws, size_t ws_size,
                              hipStream_t stream) {
    const float* emb = (const float*)d_in[0];   // [1024, 32, 256] f32
    const float* w   = (const float*)d_in[1];   // scalar
    const float* b   = (const float*)d_in[2];   // scalar
    float* out = (float*)d_out;

    float* ws = (float*)d_ws;
    float* cent    = ws;                               // 1024*256
    float* cn      = cent + N_SPK * D_EMB;             // 1024*256
    float* loo_sim = cn + N_SPK * D_EMB;               // 32768
    float* inv_e   = loo_sim + N_ROWS;                 // 32768
    float* partial = inv_e + N_ROWS;                   // 2048

    k_centroid<<<N_SPK, 256, 0, stream>>>(emb, cent, cn);
    k_rowstats<<<N_ROWS, 256, 0, stream>>>(emb, cent, loo_sim, inv_e);
    k_ge2e_main<<<N_ROWS / 16, 256, 0, stream>>>(emb, cn, loo_sim, inv_e, w, b, partial);
    k_final<<<1, 256, 0, stream>>>(partial, out);
}